// MarangoniForce_20246475833437
// MI455X (gfx1250) — compile-verified
//
#include <hip/hip_runtime.h>
#include <stdint.h>

// Marangoni force: edge-based FD gradient scatter + interface mask.
// Memory/atomic-bound (~0.4 FLOP/byte) -> no WMMA; CDNA5 async
// global->LDS copies (GVS form, NT hint) stream the edge index while
// gathers + non-returning global f32 atomics do the segment sum.

#define N_NODES 100000
#define N_EDGES 3200000

static constexpr float DSIGMA_DT  = -0.0001f;
static constexpr float PHI_THRESH = 0.3f;
static constexpr float EPS        = 1e-8f;

static constexpr int TPB   = 256;            // 8 wave32 per block
static constexpr int ITERS = 8;              // double-buffered chunks per block
static constexpr int EDGES_PER_BLOCK = TPB * ITERS;

__global__ __launch_bounds__(TPB)
void zero_acc_kernel(float4* __restrict__ acc) {
  int i = blockIdx.x * TPB + threadIdx.x;
  if (i < N_NODES) acc[i] = make_float4(0.f, 0.f, 0.f, 0.f);
}

__global__ __launch_bounds__(TPB)
void edge_scatter_kernel(const float* __restrict__ x,
                         const float* __restrict__ pos,
                         const int*   __restrict__ esrc,
                         const int*   __restrict__ edst,
                         float*       __restrict__ acc /* 4 floats per node */) {
  __shared__ int s_src[2][TPB];
  __shared__ int s_dst[2][TPB];

  const int tid = threadIdx.x;
  const long long base = (long long)blockIdx.x * EDGES_PER_BLOCK;

  // LDS byte offsets of this lane's private staging slots (generic LDS
  // pointers carry the wave-relative LDS offset in their low 32 bits).
  const unsigned lsrc[2] = { (unsigned)(unsigned long long)&s_src[0][tid],
                             (unsigned)(unsigned long long)&s_src[1][tid] };
  const unsigned ldst[2] = { (unsigned)(unsigned long long)&s_dst[0][tid],
                             (unsigned)(unsigned long long)&s_dst[1][tid] };

  // Wave-uniform 64-bit bases live in SGPRs (GVS addressing).
  const unsigned long long sb = (unsigned long long)(uintptr_t)esrc;
  const unsigned long long db = (unsigned long long)(uintptr_t)edst;

  auto issue = [&](int it, int buf) {
    long long e = base + (long long)it * TPB + tid;
    if (e < (long long)N_EDGES) {
      // Byte offset fits 32 bits (N_EDGES*4 = 12.8 MB).
      unsigned voff = (unsigned)e * 4u;
      // CDNA5 async copy global->LDS, ASYNCcnt-tracked, no VGPR return.
      // NT hint: the edge stream is read once; keep near caches for gathers.
      asm volatile("global_load_async_to_lds_b32 %0, %1, %2 th:TH_LOAD_NT"
                   :: "v"(lsrc[buf]), "v"(voff), "s"(sb) : "memory");
      asm volatile("global_load_async_to_lds_b32 %0, %1, %2 th:TH_LOAD_NT"
                   :: "v"(ldst[buf]), "v"(voff), "s"(db) : "memory");
    }
  };

  issue(0, 0);
  for (int it = 0; it < ITERS; ++it) {
    const int buf = it & 1;
    if (it + 1 < ITERS) {
      issue(it + 1, buf ^ 1);
      // 2 loads of the next chunk may remain outstanding; async loads
      // complete in order, so <=2 means the current chunk has landed.
      asm volatile("s_wait_asynccnt 0x2" ::: "memory");
    } else {
      asm volatile("s_wait_asynccnt 0x0" ::: "memory");
    }
    // Each lane reads only the LDS slot its own wave's async op filled,
    // so a per-wave ASYNCcnt wait suffices (no workgroup barrier).
    long long e = base + (long long)it * TPB + tid;
    if (e < (long long)N_EDGES) {
      const int s = s_src[buf][tid];
      const int d = s_dst[buf][tid];

      const float Ts = x[s * 9 + 3];
      const float Td = x[d * 9 + 3];
      const float dpx = pos[s * 3 + 0] - pos[d * 3 + 0];
      const float dpy = pos[s * 3 + 1] - pos[d * 3 + 1];
      const float dpz = pos[s * 3 + 2] - pos[d * 3 + 2];
      const float dist2 = dpx * dpx + dpy * dpy + dpz * dpz + EPS;
      const float w = (Ts - Td) / dist2;

      float* a = acc + (size_t)d * 4;
      atomicAdd(a + 0, w * dpx);   // global_atomic_add_f32, no return
      atomicAdd(a + 1, w * dpy);
      atomicAdd(a + 2, w * dpz);
      atomicAdd(a + 3, 1.0f);
    }
  }
}

__global__ __launch_bounds__(TPB)
void finalize_kernel(const float* __restrict__ x,
                     const float4* __restrict__ acc,
                     float* __restrict__ out) {
  int i = blockIdx.x * TPB + threadIdx.x;
  if (i < N_NODES) {
    const float4 a = acc[i];
    const float cnt = fmaxf(a.w, 1.0f);
    const float phi = x[i * 9 + 8];
    const float mask = (fabsf(phi) < PHI_THRESH) ? 1.0f : 0.0f;
    const float sc = DSIGMA_DT * mask / cnt;
    out[i * 3 + 0] = a.x * sc;
    out[i * 3 + 1] = a.y * sc;
    out[i * 3 + 2] = a.z * sc;
  }
}

extern "C" void kernel_launch(void* const* d_in, const int* in_sizes, int n_in,
                              void* d_out, int out_size, void* d_ws, size_t ws_size,
                              hipStream_t stream) {
  const float* x    = (const float*)d_in[0];   // (N_NODES, 9) f32
  const float* pos  = (const float*)d_in[1];   // (N_NODES, 3) f32
  const int*   edge = (const int*)  d_in[2];   // (2, N_EDGES) int32: row0=src, row1=dst
  float*       out  = (float*)d_out;           // (N_NODES, 3) f32
  float*       acc  = (float*)d_ws;            // N_NODES * float4 = 1.6 MB scratch

  const int nb_nodes = (N_NODES + TPB - 1) / TPB;
  const int nb_edges = (N_EDGES + EDGES_PER_BLOCK - 1) / EDGES_PER_BLOCK;

  zero_acc_kernel<<<nb_nodes, TPB, 0, stream>>>((float4*)acc);
  edge_scatter_kernel<<<nb_edges, TPB, 0, stream>>>(x, pos, edge, edge + N_EDGES, acc);
  finalize_kernel<<<nb_nodes, TPB, 0, stream>>>(x, (const float4*)acc, out);
}